// SparseGraphAttention_13718125543874
// MI455X (gfx1250) — compile-verified
//
#include <hip/hip_runtime.h>
#include <hip/hip_bf16.h>

#define ALPHA 0.2f
#define EPSV  1e-9f
#define D_IN  256
#define D_OUT 64

typedef __attribute__((ext_vector_type(2))) float v2f;
typedef __attribute__((ext_vector_type(8))) float v8f;

// ---------------------------------------------------------------------------
// Zero the output accumulator (num) and rowsum.
__global__ void zero_kernel(float* __restrict__ out, float* __restrict__ rowsum,
                            int n_out, int n_row) {
    int i = blockIdx.x * blockDim.x + threadIdx.x;
    int total = n_out + n_row;
    if (i >= total) return;
    if (i < n_out) out[i] = 0.0f;
    else           rowsum[i - n_out] = 0.0f;
}

// ---------------------------------------------------------------------------
// h = x @ W  via V_WMMA_F32_16X16X4_F32.
// One wave computes a full 16-row x 64-col strip of h (4 accumulators),
// so x is read exactly once from HBM. W is staged in LDS (64 KB),
// K-pair interleaved so each B fragment is one aligned ds_load_b64:
//   Wl[(k>>1)*128 + n*2 + (k&1)] = W[k*64 + n]
__global__ __launch_bounds__(256) void gemm_h_kernel(const float* __restrict__ x,
                                                     const float* __restrict__ W,
                                                     float* __restrict__ h,
                                                     int N) {
    __shared__ float Wl[D_IN * D_OUT];            // 64 KB of 320 KB/WGP
    for (int i = threadIdx.x; i < D_IN * D_OUT; i += 256) {
        const int k = i >> 6;        // 0..255
        const int n = i & 63;        // 0..63
        Wl[(k >> 1) * 128 + n * 2 + (k & 1)] = W[i];
    }
    __syncthreads();

    const int wave = threadIdx.x >> 5;
    const int lane = threadIdx.x & 31;
    const int rowBase = (blockIdx.x * 8 + wave) * 16;
    if (rowBase >= N) return;                     // wave-uniform: EXEC stays full

    const int m    = lane & 15;                   // M (for A) / N (for B) index
    const int half = lane >> 4;
    const int kb   = half * 2;                    // K sub-offset within step of 4

    // Clamp the load row so OOB lanes read a valid address (results are
    // discarded by the guarded store) -> no EXEC churn in the inner loop.
    const int row     = rowBase + m;
    const int rowLoad = row < N ? row : (N - 1);
    const float* xrow = x + (size_t)rowLoad * D_IN;

    v8f acc0 = {}, acc1 = {}, acc2 = {}, acc3 = {};

    for (int kk = 0; kk < D_IN; kk += 4) {
        // A fragment: A[m][kk+kb], A[m][kk+kb+1]  (8B aligned, always valid)
        const v2f a = *(const v2f*)(xrow + kk + kb);

        const int k0 = kk + kb;                   // always even
        const float* wp = &Wl[(k0 >> 1) * 128];
        // B fragments for the 4 column tiles: {B[k0][n], B[k0+1][n]}
        const v2f b0 = *(const v2f*)(wp + ( 0 + m) * 2);
        const v2f b1 = *(const v2f*)(wp + (16 + m) * 2);
        const v2f b2 = *(const v2f*)(wp + (32 + m) * 2);
        const v2f b3 = *(const v2f*)(wp + (48 + m) * 2);

        acc0 = __builtin_amdgcn_wmma_f32_16x16x4_f32(false, a, false, b0, (short)0, acc0, false, false);
        acc1 = __builtin_amdgcn_wmma_f32_16x16x4_f32(false, a, false, b1, (short)0, acc1, false, false);
        acc2 = __builtin_amdgcn_wmma_f32_16x16x4_f32(false, a, false, b2, (short)0, acc2, false, false);
        acc3 = __builtin_amdgcn_wmma_f32_16x16x4_f32(false, a, false, b3, (short)0, acc3, false, false);
    }

    // D layout: acc[ct][v] = h[rowBase + v + 8*half][ct*16 + m]
    for (int v = 0; v < 8; ++v) {
        const int r = rowBase + v + 8 * half;
        if (r < N) {
            float* hr = h + (size_t)r * D_OUT;
            hr[ 0 + m] = acc0[v];
            hr[16 + m] = acc1[v];
            hr[32 + m] = acc2[v];
            hr[48 + m] = acc3[v];
        }
    }
}

// ---------------------------------------------------------------------------
// s = h @ a_src, t = h @ a_dst  (h is L2-hot: 25.6 MB << 192 MB)
__global__ void st_kernel(const float* __restrict__ h, const float* __restrict__ att,
                          float* __restrict__ s, float* __restrict__ t, int N) {
    int i = blockIdx.x * blockDim.x + threadIdx.x;
    if (i >= N) return;
    const float* hr = h + (size_t)i * D_OUT;
    float sa = 0.0f, ta = 0.0f;
#pragma unroll
    for (int j = 0; j < D_OUT; ++j) {
        const float hv = hr[j];
        sa += hv * att[j];
        ta += hv * att[D_OUT + j];
    }
    s[i] = sa;
    t[i] = ta;
}

// ---------------------------------------------------------------------------
// Edge phase: one wave32 per edge. Gather h[dst] (L2-resident), compute
// e = exp(-leakyrelu(s[src]+t[dst])), scatter-add into num[src] / rowsum[src]
// via L2 float atomics.
__global__ __launch_bounds__(256) void edge_kernel(const int* __restrict__ ei,
                                                   const float* __restrict__ h,
                                                   const float* __restrict__ s,
                                                   const float* __restrict__ t,
                                                   float* __restrict__ num,
                                                   float* __restrict__ rowsum,
                                                   int E) {
    const int gtid = blockIdx.x * blockDim.x + threadIdx.x;
    const int e    = gtid >> 5;
    const int lane = gtid & 31;
    if (e >= E) return;

    const int src = ei[e];
    const int dst = ei[E + e];

    const float z  = s[src] + t[dst];            // same addr across wave -> broadcast
    const float lr = z > 0.0f ? z : ALPHA * z;
    const float ev = __expf(-lr);

    if (lane == 0) atomicAdd(rowsum + src, ev);

    const float* hd = h + (size_t)dst * D_OUT;
    float*       nr = num + (size_t)src * D_OUT;
    atomicAdd(nr + lane,      ev * hd[lane]);
    atomicAdd(nr + lane + 32, ev * hd[lane + 32]);
}

// ---------------------------------------------------------------------------
// h_prime = num / (rowsum + eps), in place on d_out.
__global__ void finalize_kernel(float* __restrict__ out, const float* __restrict__ rowsum,
                                int N) {
    const int idx = blockIdx.x * blockDim.x + threadIdx.x;
    if (idx >= N * D_OUT) return;
    const int i = idx >> 6;
    out[idx] = out[idx] / (rowsum[i] + EPSV);
}

// ---------------------------------------------------------------------------
extern "C" void kernel_launch(void* const* d_in, const int* in_sizes, int n_in,
                              void* d_out, int out_size, void* d_ws, size_t ws_size,
                              hipStream_t stream) {
    const float* x   = (const float*)d_in[0];
    const int*   ei  = (const int*)  d_in[1];
    const float* W   = (const float*)d_in[2];
    const float* att = (const float*)d_in[3];
    float*       out = (float*)d_out;

    const int N = in_sizes[0] / D_IN;   // 100000
    const int E = in_sizes[1] / 2;      // 1600000

    // Workspace layout (all L2-resident): h | rowsum | s | t
    float* h      = (float*)d_ws;
    float* rowsum = h + (size_t)N * D_OUT;
    float* s      = rowsum + N;
    float* t      = s + N;

    const int totalZero = N * D_OUT + N;
    zero_kernel<<<(totalZero + 255) / 256, 256, 0, stream>>>(out, rowsum, N * D_OUT, N);

    const int nRowTiles = (N + 15) / 16;
    const int nBlocks   = (nRowTiles + 7) / 8;   // 8 waves/block, 1 strip/wave
    gemm_h_kernel<<<nBlocks, 256, 0, stream>>>(x, W, h, N);

    st_kernel<<<(N + 255) / 256, 256, 0, stream>>>(h, att, s, t, N);

    const long long ethreads = (long long)E * 32;
    edge_kernel<<<(int)((ethreads + 255) / 256), 256, 0, stream>>>(ei, h, s, t, out, rowsum, E);

    finalize_kernel<<<(N * D_OUT + 255) / 256, 256, 0, stream>>>(out, rowsum, N);
}